// ADL_26611617366422
// MI455X (gfx1250) — compile-verified
//
#include <hip/hip_runtime.h>

// ADL forward: attn = sigmoid(1x1conv(fm)), drop top-M spatial locations, mask fm.
// B=64, C=1024, H*W=1024, M=768.
// d_out = [dropped: B*C*HW floats][attn: B*HW floats]
//
// Memory plan (MI455X: 23.3 TB/s HBM, 192 MB L2):
//  - fm is 256 MB; per-batch slice fm[b] is 4 MB. Kernel 1 streams fm in
//    ascending order (default RT caching -> populates L2; at the end L2 holds
//    the last ~192 MB). Kernel 3 re-reads fm in DESCENDING block order so its
//    first reads hit the most-recently-resident lines (~75% L2 hit instead of
//    LRU thrash), and uses NT hints on its own load/store so the 256 MB output
//    stream does not evict the fm lines it is about to consume.

typedef __attribute__((ext_vector_type(2))) float v2f;
typedef __attribute__((ext_vector_type(4))) float v4f;
typedef __attribute__((ext_vector_type(8))) float v8f;

#define ADL_B  64
#define ADL_C  1024
#define ADL_HW 1024

// ---------------------------------------------------------------------------
// Kernel 1: attn[b,s] = sigmoid(sum_c fm[b,c,s]*w[c] + bias) via
// V_WMMA_F32_16X16X4_F32 (the only fp32 WMMA shape -> bit-comparable to the
// reference fp32 einsum, which matters because top-k ranking is
// order-sensitive). One wave owns 16 spatial positions (M); B broadcasts
// w[c..c+3] across all 16 N columns; K=1024 accumulated in steps of 4.
// ---------------------------------------------------------------------------
__global__ __launch_bounds__(256) void adl_attn_wmma(
    const float* __restrict__ fm, const float* __restrict__ w,
    const float* __restrict__ bias, float* __restrict__ attn) {
  const int wave = (blockIdx.x << 3) + (threadIdx.x >> 5);  // 4096 waves
  const int b    = wave >> 6;          // 64 tiles of 16 spatial per batch
  const int s0   = (wave & 63) << 4;   // spatial base of this 16-wide tile
  const int lane = threadIdx.x & 31;
  const int m    = lane & 15;          // spatial row within tile (A: M index)
  const int kg   = (lane >> 4) << 1;   // K sub-group: 0 for lanes<16, 2 else

  const float* __restrict__ fmb = fm + (size_t)b * ADL_C * ADL_HW;

  v8f acc = {};
  for (int c = 0; c < ADL_C; c += 4) {
    v2f a, bv;
    a.x  = fmb[(size_t)(c + kg    ) * ADL_HW + s0 + m];
    a.y  = fmb[(size_t)(c + kg + 1) * ADL_HW + s0 + m];
    bv.x = w[c + kg];
    bv.y = w[c + kg + 1];
    acc = __builtin_amdgcn_wmma_f32_16x16x4_f32(
        /*neg_a=*/false, a, /*neg_b=*/false, bv,
        /*c_mod=*/(short)0, acc, /*reuse_a=*/false, /*reuse_b=*/false);
  }

  // Column N=0 lives in lanes 0 (M=0..7) and 16 (M=8..15).
  if (m == 0) {
    const float bb = bias[0];
    const int mb = (lane >> 4) << 3;
#pragma unroll
    for (int r = 0; r < 8; ++r) {
      float x = acc[r] + bb;
      attn[b * ADL_HW + s0 + mb + r] = 1.0f / (1.0f + __expf(-x));
    }
  }
}

// ---------------------------------------------------------------------------
// Kernel 2: per-batch top-M drop mask via exhaustive descending-rank in LDS.
// rank = #(a_j > a_i) + #(a_j == a_i && j < i)  (matches jax top_k tie-break)
// keep (mask=1) iff rank >= M. 64 blocks x 1M LDS compares: negligible.
// ---------------------------------------------------------------------------
__global__ __launch_bounds__(1024) void adl_mask(
    const float* __restrict__ attn, const int* __restrict__ Mp,
    float* __restrict__ mask) {
  __shared__ float sa[ADL_HW];
  const int b = blockIdx.x;
  const int t = threadIdx.x;
  const float v = attn[b * ADL_HW + t];
  sa[t] = v;
  __syncthreads();
  const int M = *Mp;
  int rank = 0;
  for (int j = 0; j < ADL_HW; ++j) {
    const float o = sa[j];
    rank += (o > v) | ((o == v) & (j < t));
  }
  mask[b * ADL_HW + t] = (rank < M) ? 0.0f : 1.0f;
}

// ---------------------------------------------------------------------------
// Kernel 3: dropped[b,c,s] = fm[b,c,s] * mask[b,s], float4-vectorized (b128).
// Blocks are issued in DESCENDING address order to pick up kernel 1's L2
// residency (MRU-first instead of LRU-thrash). fm load is non-temporal
// (last use: hit L2 then release the line); output store is non-temporal
// (pure 256 MB stream, never re-read -> keep it out of L2).
// ---------------------------------------------------------------------------
__global__ __launch_bounds__(256) void adl_apply(
    const float* __restrict__ fm, const float* __restrict__ mask,
    float* __restrict__ out) {
  // Reverse block issue order; keep lanes ascending within a block for
  // coalescing.
  const size_t rb = (size_t)(gridDim.x - 1 - blockIdx.x);
  const size_t i4 = rb * blockDim.x + threadIdx.x;
  const size_t e  = i4 << 2;                 // element base, multiple of 4
  const size_t b  = e >> 20;                 // / (C*HW)
  const size_t s  = e & (ADL_HW - 1);        // spatial offset (mult of 4)

  const v4f f  = __builtin_nontemporal_load((const v4f*)(fm + e));
  const v4f mk = *(const v4f*)(mask + (b << 10) + s);  // tiny, keep cached
  const v4f o  = f * mk;
  __builtin_nontemporal_store(o, (v4f*)(out + e));
}

extern "C" void kernel_launch(void* const* d_in, const int* in_sizes, int n_in,
                              void* d_out, int out_size, void* d_ws, size_t ws_size,
                              hipStream_t stream) {
  const float* fm  = (const float*)d_in[0];  // [64,1024,32,32] f32
  const float* w   = (const float*)d_in[1];  // [1024] f32
  const float* cb  = (const float*)d_in[2];  // [1] f32
  const int*   Mp  = (const int*)d_in[3];    // scalar int (768)

  float* out      = (float*)d_out;                               // dropped
  float* attn_out = out + (size_t)ADL_B * ADL_C * ADL_HW;        // attn tail
  float* mask     = (float*)d_ws;                                // 256 KB

  // 1) fp32 WMMA channel contraction -> attn (written straight to d_out tail)
  adl_attn_wmma<<<512, 256, 0, stream>>>(fm, w, cb, attn_out);
  // 2) per-batch top-M drop mask
  adl_mask<<<ADL_B, 1024, 0, stream>>>(attn_out, Mp, mask);
  // 3) broadcast masked multiply (traffic-dominant, float4, reverse order, NT)
  const int total4 = (ADL_B * ADL_C * ADL_HW) / 4;  // 16,777,216
  adl_apply<<<total4 / 256, 256, 0, stream>>>(fm, mask, out);
}